// DeformationModel_80444737454424
// MI455X (gfx1250) — compile-verified
//
#include <hip/hip_runtime.h>
#include <math.h>

// MI455X / gfx1250. Memory-bound streaming kernel:
//   302 MB one-pass @ 23.3 TB/s  =>  ~13 us roofline.
// Strategy: 4 points per thread, all I/O as 128-bit non-temporal
// global loads/stores (th:TH_NT), wave32-friendly 256-thread blocks.

typedef float v4f __attribute__((ext_vector_type(4)));

#define EPS 1e-6f

__device__ __forceinline__ void deform_point(float x0, float x1, float x2,
                                             const float* __restrict__ q,
                                             float* __restrict__ o) {
    float wx = q[0], wy = q[1], wz = q[2];
    float vx = q[3], vy = q[4], vz = q[5];
    float px = q[6], py = q[7], pz = q[8];
    float tx = q[9], ty = q[10], tz = q[11];

    float theta = sqrtf(wx * wx + wy * wy + wz * wz) + EPS;
    float inv = 1.0f / theta;
    wx *= inv; wy *= inv; wz *= inv;
    vx *= inv; vy *= inv; vz *= inv;

    float s, c;
    sincosf(theta, &s, &c);
    float c1 = 1.0f - c;       // (1 - cos)
    float ts = theta - s;      // (theta - sin)

    // u = x + pivot
    float ux = x0 + px, uy = x1 + py, uz = x2 + pz;
    // K u = w x u
    float cux = wy * uz - wz * uy;
    float cuy = wz * ux - wx * uz;
    float cuz = wx * uy - wy * ux;
    // K^2 u = w x (w x u)
    float dux = wy * cuz - wz * cuy;
    float duy = wz * cux - wx * cuz;
    float duz = wx * cuy - wy * cux;
    // R u = u + s*Ku + c1*K^2 u
    float rx = ux + s * cux + c1 * dux;
    float ry = uy + s * cuy + c1 * duy;
    float rz = uz + s * cuz + c1 * duz;

    // K v, K^2 v
    float cvx = wy * vz - wz * vy;
    float cvy = wz * vx - wx * vz;
    float cvz = wx * vy - wy * vx;
    float dvx = wy * cvz - wz * cvy;
    float dvy = wz * cvx - wx * cvz;
    float dvz = wx * cvy - wy * cvx;
    // p = theta*v + c1*Kv + (theta - s)*K^2 v
    float ppx = theta * vx + c1 * cvx + ts * dvx;
    float ppy = theta * vy + c1 * cvy + ts * dvy;
    float ppz = theta * vz + c1 * cvz + ts * dvz;

    // out = R(x+pivot) + p - pivot + t - x
    o[0] = rx + ppx - px + tx - x0;
    o[1] = ry + ppy - py + ty - x1;
    o[2] = rz + ppz - pz + tz - x2;
}

__global__ __launch_bounds__(256) void deform_kernel(
    const float* __restrict__ pos,   // (N,3)
    const float* __restrict__ net,   // (N,12)
    float* __restrict__ out,         // (N,3)
    int npts) {
    const long tid = (long)blockIdx.x * 256 + threadIdx.x;
    const long p0 = tid * 4;  // first point of this thread's quad
    if (p0 >= npts) return;

    if (p0 + 4 <= npts) {
        // ---- fast path: 4 points, all 16B-aligned 128-bit NT traffic ----
        const v4f* pp = (const v4f*)(pos + p0 * 3);   // 48 B  = 3 x v4f
        const v4f* np = (const v4f*)(net + p0 * 12);  // 192 B = 12 x v4f

        float X[12];
        float Y[48];
#pragma unroll
        for (int i = 0; i < 3; ++i) {
            v4f t = __builtin_nontemporal_load(pp + i);
            X[4 * i + 0] = t.x; X[4 * i + 1] = t.y;
            X[4 * i + 2] = t.z; X[4 * i + 3] = t.w;
        }
#pragma unroll
        for (int i = 0; i < 12; ++i) {
            v4f t = __builtin_nontemporal_load(np + i);
            Y[4 * i + 0] = t.x; Y[4 * i + 1] = t.y;
            Y[4 * i + 2] = t.z; Y[4 * i + 3] = t.w;
        }

        float O[12];
#pragma unroll
        for (int j = 0; j < 4; ++j) {
            deform_point(X[3 * j + 0], X[3 * j + 1], X[3 * j + 2],
                         &Y[12 * j], &O[3 * j]);
        }

        v4f* op = (v4f*)(out + p0 * 3);
#pragma unroll
        for (int i = 0; i < 3; ++i) {
            v4f t;
            t.x = O[4 * i + 0]; t.y = O[4 * i + 1];
            t.z = O[4 * i + 2]; t.w = O[4 * i + 3];
            __builtin_nontemporal_store(t, op + i);
        }
    } else {
        // ---- tail path (never taken for N = 4M, kept for generality) ----
        for (long p = p0; p < npts; ++p) {
            float q[12];
#pragma unroll
            for (int i = 0; i < 12; ++i) q[i] = net[p * 12 + i];
            float o[3];
            deform_point(pos[p * 3 + 0], pos[p * 3 + 1], pos[p * 3 + 2], q, o);
            out[p * 3 + 0] = o[0];
            out[p * 3 + 1] = o[1];
            out[p * 3 + 2] = o[2];
        }
    }
}

extern "C" void kernel_launch(void* const* d_in, const int* in_sizes, int n_in,
                              void* d_out, int out_size, void* d_ws, size_t ws_size,
                              hipStream_t stream) {
    const float* pos = (const float*)d_in[0];  // undeformed_positions (N,3)
    const float* net = (const float*)d_in[1];  // network_output      (N,12)
    float* out = (float*)d_out;                // unmasked_offsets    (N,3)

    const int npts = in_sizes[0] / 3;
    const int nquads = (npts + 3) / 4;
    const int threads = 256;                   // 8 wave32 waves per block
    const int blocks = (nquads + threads - 1) / threads;

    deform_kernel<<<blocks, threads, 0, stream>>>(pos, net, out, npts);
}